// Liftnet_5471788335319
// MI455X (gfx1250) — compile-verified
//
#include <hip/hip_runtime.h>

// ---------------------------------------------------------------------------
// Problem constants (from reference)
// ---------------------------------------------------------------------------
#define S_CAMS  4
#define C_FEAT  96
#define D_DEPTH 32
#define HF      45
#define WF      80
#define YV      128
#define ZV      8
#define XV      128
#define HW      (HF * WF)            // 3600
#define EPSF    1e-6f

typedef float v2f __attribute__((ext_vector_type(2)));
typedef float v8f __attribute__((ext_vector_type(8)));

// Workspace layout (bytes):
//   depthT : [S][HF][WF][D]  = 460800 floats  @ 0
//   featT  : [S][HF][WF][C]  = 1382400 floats @ 1843200
//   Amat   : [16][4]         = 64 floats      @ 7372800
#define WS_DEPTH_OFF 0
#define WS_FEAT_OFF  (460800u * 4u)
#define WS_AMAT_OFF  (WS_FEAT_OFF + 1382400u * 4u)

// ---------------------------------------------------------------------------
// Setup kernel 1: build the stacked 16x4 projection matrix.
// Row m = 4*s + q :  q=0,1,2 -> (K[s]@E[s]) rows 0,1,2 ;  q=3 -> E[s] row 2
// ---------------------------------------------------------------------------
__global__ void build_amat_kernel(const float* __restrict__ K,
                                  const float* __restrict__ E,
                                  float* __restrict__ Amat) {
  int t = threadIdx.x;
  if (t >= 16) return;
  int s = t >> 2, q = t & 3;
  const float* Ks = K + s * 16;
  const float* Es = E + s * 16;
  float r0, r1, r2, r3;
  if (q == 3) {
    r0 = Es[8]; r1 = Es[9]; r2 = Es[10]; r3 = Es[11];
  } else {
    float k0 = Ks[q * 4 + 0], k1 = Ks[q * 4 + 1], k2 = Ks[q * 4 + 2], k3 = Ks[q * 4 + 3];
    r0 = k0 * Es[0] + k1 * Es[4] + k2 * Es[8]  + k3 * Es[12];
    r1 = k0 * Es[1] + k1 * Es[5] + k2 * Es[9]  + k3 * Es[13];
    r2 = k0 * Es[2] + k1 * Es[6] + k2 * Es[10] + k3 * Es[14];
    r3 = k0 * Es[3] + k1 * Es[7] + k2 * Es[11] + k3 * Es[15];
  }
  Amat[t * 4 + 0] = r0;
  Amat[t * 4 + 1] = r1;
  Amat[t * 4 + 2] = r2;
  Amat[t * 4 + 3] = r3;
}

// ---------------------------------------------------------------------------
// Setup kernel 2: softmax over depth channels, output depth-contiguous layout
// depthT[((s*HF+h)*WF+w)*32 + d]
// ---------------------------------------------------------------------------
__global__ void softmax_depth_kernel(const float* __restrict__ in,
                                     float* __restrict__ depthT) {
  int t = blockIdx.x * blockDim.x + threadIdx.x;
  if (t >= S_CAMS * HW) return;
  int s  = t / HW;
  int hw = t - s * HW;
  const float* p = in + (size_t)s * (D_DEPTH + C_FEAT) * HW + hw;
  float v[D_DEPTH];
  float m = -3.402823466e38f;
#pragma unroll
  for (int d = 0; d < D_DEPTH; ++d) { v[d] = p[d * HW]; m = fmaxf(m, v[d]); }
  float sum = 0.0f;
#pragma unroll
  for (int d = 0; d < D_DEPTH; ++d) { v[d] = expf(v[d] - m); sum += v[d]; }
  float inv = 1.0f / sum;
  float* o = depthT + (size_t)t * D_DEPTH;
#pragma unroll
  for (int d = 0; d < D_DEPTH; ++d) o[d] = v[d] * inv;
}

// ---------------------------------------------------------------------------
// Setup kernel 3: transpose features to channel-contiguous layout
// featT[((s*HF+h)*WF+w)*96 + c] = feat_cams[s, 32+c, h, w]
// ---------------------------------------------------------------------------
__global__ void transpose_feat_kernel(const float* __restrict__ in,
                                      float* __restrict__ featT) {
  int t = blockIdx.x * blockDim.x + threadIdx.x;
  if (t >= S_CAMS * HW * C_FEAT) return;
  int s   = t / (HW * C_FEAT);
  int rem = t - s * (HW * C_FEAT);
  int hw  = rem / C_FEAT;
  int c   = rem - hw * C_FEAT;
  featT[t] = in[((size_t)(s * (D_DEPTH + C_FEAT) + D_DEPTH + c)) * HW + hw];
}

// ---------------------------------------------------------------------------
// Main kernel. One wave = 16 voxels (fixed y,z; 16 consecutive x).
// WMMA f32 16x16x4 computes Proj(16x16) = Amat(16x4) x Pts(4x16):
//   row m = 4*s + {px, py, pz, zcam},  col n = voxel.
// Lane pair (l, l+16) serves voxel l%16; lane owns channels [half*48, +48).
// ---------------------------------------------------------------------------
__global__ __launch_bounds__(256) void lift_kernel(
    const float* __restrict__ depthT,
    const float* __restrict__ featT,
    const float* __restrict__ Amat,
    float* __restrict__ out) {
  const int lane = threadIdx.x & 31;
  const int col  = lane & 15;
  const int half = lane >> 4;
  const int wave = blockIdx.x * (blockDim.x >> 5) + (threadIdx.x >> 5);

  // voxel tile: p0 = y*(Z*X) + z*X + x0, 16 consecutive x
  const int p0 = wave << 4;
  const int y  = p0 >> 10;        // / (ZV*XV) = /1024
  const int z  = (p0 >> 7) & 7;
  const int x0 = p0 & 127;
  const int xv = x0 + col;

  const float xw = -25.0f + ((float)xv + 0.5f) * (50.0f / 128.0f);
  const float yw = -25.0f + ((float)y  + 0.5f) * (50.0f / 128.0f);
  const float zw = ((float)z + 0.5f) * 0.5f;

  // A-matrix operand: lanes 0-15 -> row=lane, K=0,1 ; lanes 16-31 -> row=lane-16, K=2,3
  const float* ap = Amat + col * 4 + (half ? 2 : 0);
  v2f a; a.x = ap[0]; a.y = ap[1];
  // B-matrix operand (4x16): lanes 0-15 hold K=0,1 ; lanes 16-31 hold K=2,3 ; col = lane%16
  v2f b; b.x = half ? zw : xw;
         b.y = half ? 1.0f : yw;
  v8f cz = {0.f, 0.f, 0.f, 0.f, 0.f, 0.f, 0.f, 0.f};
  v8f dm = __builtin_amdgcn_wmma_f32_16x16x4_f32(
      /*neg_a=*/false, a, /*neg_b=*/false, b,
      /*c_mod=*/(short)0, cz, /*reuse_a=*/false, /*reuse_b=*/false);

  // Gather all 16 projection rows into each lane (half-wave exchange).
  float pr[16];
#pragma unroll
  for (int r = 0; r < 8; ++r) {
    float own = dm[r];                       // lanes<16: row r ; lanes>=16: row 8+r
    float oth = __shfl_xor(own, 16, 32);
    pr[r]     = half ? oth : own;
    pr[r + 8] = half ? own : oth;
  }

  float acc[48];
#pragma unroll
  for (int c = 0; c < 48; ++c) acc[c] = 0.0f;
  float den = 0.0f;
  const int c0 = half * 48;

#pragma unroll
  for (int s = 0; s < S_CAMS; ++s) {
    const float px = pr[4 * s + 0], py = pr[4 * s + 1];
    const float pz = pr[4 * s + 2], zcam = pr[4 * s + 3];
    const float zcl = fmaxf(pz, EPSF);
    const float xp = px / zcl, yp = py / zcl;
    const float zt = 31.0f * (zcam - 2.0f) * (1.0f / 30.0f);
    const bool valid = (xp > -0.5f) && (xp < (float)WF - 0.5f) &&
                       (yp > -0.5f) && (yp < (float)HF - 0.5f) && (zcam > 0.0f);
    // grid_sample-style coordinate remap
    const float xs = xp * ((float)WF / (WF - 1.0f)) - 0.5f;
    const float ys = yp * ((float)HF / (HF - 1.0f)) - 0.5f;
    const float zs = zt * ((float)D_DEPTH / (D_DEPTH - 1.0f)) - 0.5f;
    const float xf = floorf(xs), yf = floorf(ys), zf = floorf(zs);
    const int ix0 = (int)xf, iy0 = (int)yf, iz0 = (int)zf;
    const float fx = xs - xf, fy = ys - yf, fz = zs - zf;
    const float wx[2] = {1.0f - fx, fx};
    const float wy[2] = {1.0f - fy, fy};
    const float wz0 = (1.0f - fz) * ((iz0 >= 0 && iz0 < D_DEPTH) ? 1.0f : 0.0f);
    const float wz1 = fz * ((iz0 + 1 >= 0 && iz0 + 1 < D_DEPTH) ? 1.0f : 0.0f);
    const int z0c = min(max(iz0, 0), D_DEPTH - 1);
    const int z1c = min(max(iz0 + 1, 0), D_DEPTH - 1);

    float coef[4];
    int   fbase[4];
    float ov = 0.0f;
#pragma unroll
    for (int j = 0; j < 4; ++j) {
      const int dx = j & 1, dy = j >> 1;
      const int xi = ix0 + dx, yi = iy0 + dy;
      const bool inb = (xi >= 0) && (xi < WF) && (yi >= 0) && (yi < HF);
      const float wxy = wx[dx] * wy[dy];
      float cf = 0.0f;
      int  fb = 0;
      if (inb) {
        const int hwb = (s * HF + yi) * WF + xi;
        const float* dp = depthT + (hwb << 5);
        const float zpart = wz0 * dp[z0c] + wz1 * dp[z1c];
        cf = wxy * zpart;
        fb = hwb * C_FEAT + c0;
        ov += wxy * (wz0 + wz1);
      }
      coef[j] = cf;
      fbase[j] = fb;
    }

    const float onev = fmaxf(valid ? ov : 0.0f, 1.0f);
    den += onev;

    if (valid) {
#pragma unroll
      for (int j = 0; j < 4; ++j) {
        const float cf = coef[j] * onev;
        if (cf != 0.0f) {
          const float4* fp = (const float4*)(featT + fbase[j]);
#pragma unroll
          for (int q = 0; q < 12; ++q) {
            float4 v = fp[q];
            acc[q * 4 + 0] = fmaf(cf, v.x, acc[q * 4 + 0]);
            acc[q * 4 + 1] = fmaf(cf, v.y, acc[q * 4 + 1]);
            acc[q * 4 + 2] = fmaf(cf, v.z, acc[q * 4 + 2]);
            acc[q * 4 + 3] = fmaf(cf, v.w, acc[q * 4 + 3]);
          }
        }
      }
    }
  }

  const float inv = 1.0f / (EPSF + den);
  // out[b, c*Z + z, y, x]  with (Y,X) = (128,128)
#pragma unroll
  for (int c = 0; c < 48; ++c) {
    const int cg = c0 + c;
    out[(((cg * ZV + z) * YV + y) << 7) + xv] = acc[c] * inv;
  }
}

// ---------------------------------------------------------------------------
// Launch
// ---------------------------------------------------------------------------
extern "C" void kernel_launch(void* const* d_in, const int* in_sizes, int n_in,
                              void* d_out, int out_size, void* d_ws, size_t ws_size,
                              hipStream_t stream) {
  (void)in_sizes; (void)n_in; (void)out_size; (void)ws_size;
  const float* feat_cams = (const float*)d_in[0];
  const float* Kmat      = (const float*)d_in[1];
  const float* Emat      = (const float*)d_in[2];
  float* out = (float*)d_out;
  char*  ws  = (char*)d_ws;
  float* depthT = (float*)(ws + WS_DEPTH_OFF);
  float* featT  = (float*)(ws + WS_FEAT_OFF);
  float* Amat   = (float*)(ws + WS_AMAT_OFF);

  build_amat_kernel<<<1, 16, 0, stream>>>(Kmat, Emat, Amat);

  const int nsm = S_CAMS * HW;                          // 14400
  softmax_depth_kernel<<<(nsm + 255) / 256, 256, 0, stream>>>(feat_cams, depthT);

  const int ntr = S_CAMS * HW * C_FEAT;                 // 1382400
  transpose_feat_kernel<<<(ntr + 255) / 256, 256, 0, stream>>>(feat_cams, featT);

  // 8192 voxel tiles of 16 voxels; 8 waves per 256-thread block -> 1024 blocks
  lift_kernel<<<1024, 256, 0, stream>>>(depthT, featT, Amat, out);
}